// AutoDim_75153337745779
// MI455X (gfx1250) — compile-verified
//
#include <hip/hip_runtime.h>

typedef float v2f __attribute__((ext_vector_type(2)));
typedef float v8f __attribute__((ext_vector_type(8)));

#define NB 16384
#define NF 39
#define NE 32
#define ROW_STRIDE (NF * NE)   // 1248 floats per b-row
#define INV_N (1.0f / (float)(NB * NF))

__device__ __forceinline__ v8f wmma4(v2f a, v2f b, v8f c) {
  // V_WMMA_F32_16X16X4_F32 : D(16x16,f32) = A(16x4,f32) * B(4x16,f32) + C
  return __builtin_amdgcn_wmma_f32_16x16x4_f32(false, a, false, b, (short)0, c,
                                               false, false);
}

// B-fragment (4x16 KxN, f32): lane n = lane&15 -> column, h = lane>>4.
// VGPR v holds K = v + 2*h. Weight w is [.., i, e] row-major with E stride 32.
__device__ __forceinline__ v2f load_bfrag(const float* __restrict__ w, int i0,
                                          int e) {
  v2f b;
  b.x = w[i0 * NE + e];
  b.y = w[(i0 + 1) * NE + e];
  return b;
}

// ---------------------------------------------------------------------------
// Kernel 0: zero the 256 stats accumulators in workspace
// ---------------------------------------------------------------------------
__global__ void k_zero(float* __restrict__ p) { p[threadIdx.x] = 0.0f; }

// ---------------------------------------------------------------------------
// Kernel 1: stats pass. For each (f, b-tile) compute y_k tiles with WMMA and
// accumulate per-(k,e) sum and sum-of-squares into ws[0..127] / ws[128..255].
// ---------------------------------------------------------------------------
__global__ __launch_bounds__(256) void k_stats(
    const float* __restrict__ emb, const float* __restrict__ w4,
    const float* __restrict__ w8, const float* __restrict__ w16,
    const float* __restrict__ w32, float* __restrict__ sums) {
  __shared__ float lds_sum[4 * NE];
  __shared__ float lds_sq[4 * NE];
  const int tid = threadIdx.x;
  if (tid < 128) {
    lds_sum[tid] = 0.0f;
    lds_sq[tid] = 0.0f;
  }
  __syncthreads();

  const int f = blockIdx.x;
  const int wave = tid >> 5;
  const int lane = tid & 31;
  const int n = lane & 15;   // column (e mod 16) / row-in-tile for A
  const int h = lane >> 4;   // lane half -> K pair select

  // Preload all B fragments for this field (held in VGPRs across the loop).
  const float* pw32 = w32 + f * 32 * NE;
  const float* pw16 = w16 + f * 16 * NE;
  const float* pw8 = w8 + f * 8 * NE;
  const float* pw4 = w4 + f * 4 * NE;
  v2f b32f[8][2], b16f[4][2], b8f[2][2], b4f[1][2];
#pragma unroll
  for (int s = 0; s < 8; ++s)
#pragma unroll
    for (int nh = 0; nh < 2; ++nh)
      b32f[s][nh] = load_bfrag(pw32, 4 * s + 2 * h, n + 16 * nh);
#pragma unroll
  for (int s = 0; s < 4; ++s)
#pragma unroll
    for (int nh = 0; nh < 2; ++nh)
      b16f[s][nh] = load_bfrag(pw16, 4 * s + 2 * h, n + 16 * nh);
#pragma unroll
  for (int s = 0; s < 2; ++s)
#pragma unroll
    for (int nh = 0; nh < 2; ++nh)
      b8f[s][nh] = load_bfrag(pw8, 4 * s + 2 * h, n + 16 * nh);
#pragma unroll
  for (int nh = 0; nh < 2; ++nh)
    b4f[0][nh] = load_bfrag(pw4, 2 * h, n + 16 * nh);

  float sacc[4][2] = {};
  float qacc[4][2] = {};
  const int kb = 2 * h;  // A-fragment K base for this lane half

  for (int t = 0; t < 8; ++t) {
    const int btile = blockIdx.y * 8 + wave + 128 * t;
    const float* prow = emb + (size_t)(btile * 16 + n) * ROW_STRIDE + f * NE;
    if (t < 7) __builtin_prefetch(prow + (size_t)128 * 16 * ROW_STRIDE, 0, 3);
    v2f a[8];
#pragma unroll
    for (int s = 0; s < 8; ++s)
      a[s] = *reinterpret_cast<const v2f*>(prow + 4 * s + kb);

#pragma unroll
    for (int k = 0; k < 4; ++k) {
      const int nslice = 1 << k;  // 1,2,4,8 k-slices (d = 4,8,16,32)
      v8f c0 = {};
      v8f c1 = {};
#pragma unroll
      for (int s = 0; s < 8; ++s) {
        if (s < nslice) {
          v2f bf0, bf1;
          if (k == 3) { bf0 = b32f[s][0]; bf1 = b32f[s][1]; }
          else if (k == 2) { bf0 = b16f[s & 3][0]; bf1 = b16f[s & 3][1]; }
          else if (k == 1) { bf0 = b8f[s & 1][0]; bf1 = b8f[s & 1][1]; }
          else { bf0 = b4f[0][0]; bf1 = b4f[0][1]; }
          c0 = wmma4(a[s], bf0, c0);
          c1 = wmma4(a[s], bf1, c1);
        }
      }
#pragma unroll
      for (int r = 0; r < 8; ++r) {
        float v0 = c0[r], v1 = c1[r];
        sacc[k][0] += v0;
        qacc[k][0] += v0 * v0;
        sacc[k][1] += v1;
        qacc[k][1] += v1 * v1;
      }
    }
  }

  // Reduce within the block via LDS float atomics, then one global atomic
  // per LDS slot.
#pragma unroll
  for (int k = 0; k < 4; ++k)
#pragma unroll
    for (int nh = 0; nh < 2; ++nh) {
      atomicAdd(&lds_sum[k * NE + n + 16 * nh], sacc[k][nh]);
      atomicAdd(&lds_sq[k * NE + n + 16 * nh], qacc[k][nh]);
    }
  __syncthreads();
  if (tid < 128) {
    atomicAdd(&sums[tid], lds_sum[tid]);
    atomicAdd(&sums[128 + tid], lds_sq[tid]);
  }
}

// ---------------------------------------------------------------------------
// Kernel 2: finalize stats -> mu, rsqrt(var+eps); Gumbel softmax gates;
// build fused weight W'[f][i][e] and bias[f][e].
// ---------------------------------------------------------------------------
__global__ __launch_bounds__(256) void k_finalize(
    const float* __restrict__ sums, const float* __restrict__ gate,
    const float* __restrict__ noise, const float* __restrict__ w4,
    const float* __restrict__ w8, const float* __restrict__ w16,
    const float* __restrict__ w32, float* __restrict__ Wf,
    float* __restrict__ biasF) {
  __shared__ float rsL[128];
  __shared__ float muL[128];
  __shared__ float gsL[NF * 4];
  const int tid = threadIdx.x;
  if (tid < 128) {
    float mu = sums[tid] * INV_N;
    float var = sums[128 + tid] * INV_N - mu * mu;
    muL[tid] = mu;
    rsL[tid] = rsqrtf(var + 1e-5f);
  }
  if (tid < NF) {
    float lg[4];
    float mx = -1e30f;
#pragma unroll
    for (int k = 0; k < 4; ++k) {
      float u = noise[tid * 4 + k];
      float g = -logf(-logf(u + 1e-10f) + 1e-10f);
      lg[k] = gate[tid * 4 + k] + g;  // tau = 1.0
      mx = fmaxf(mx, lg[k]);
    }
    float den = 0.0f;
#pragma unroll
    for (int k = 0; k < 4; ++k) {
      lg[k] = expf(lg[k] - mx);
      den += lg[k];
    }
#pragma unroll
    for (int k = 0; k < 4; ++k) gsL[tid * 4 + k] = lg[k] / den;
  }
  __syncthreads();

  for (int idx = blockIdx.x * 256 + tid; idx < NF * 32 * 32;
       idx += gridDim.x * 256) {
    const int f = idx >> 10;
    const int r = idx & 1023;  // i*32 + e
    const int i = r >> 5;
    const int e = r & 31;
    float acc = gsL[f * 4 + 3] * rsL[96 + e] * w32[f * 1024 + r];
    if (i < 16) acc += gsL[f * 4 + 2] * rsL[64 + e] * w16[f * 512 + r];
    if (i < 8) acc += gsL[f * 4 + 1] * rsL[32 + e] * w8[f * 256 + r];
    if (i < 4) acc += gsL[f * 4 + 0] * rsL[e] * w4[f * 128 + r];
    Wf[idx] = 0.25f * acc;
  }
  for (int idx = blockIdx.x * 256 + tid; idx < NF * 32;
       idx += gridDim.x * 256) {
    const int f = idx >> 5;
    const int e = idx & 31;
    float b = 0.0f;
#pragma unroll
    for (int k = 0; k < 4; ++k)
      b += gsL[f * 4 + k] * rsL[k * 32 + e] * muL[k * 32 + e];
    biasF[idx] = -0.25f * b;
  }
}

// ---------------------------------------------------------------------------
// Kernel 3: fused output GEMM: out[b,f,:] = emb[b,f,:] @ W'[f] + bias[f]
// ---------------------------------------------------------------------------
__global__ __launch_bounds__(256) void k_out(const float* __restrict__ emb,
                                             const float* __restrict__ Wf,
                                             const float* __restrict__ biasF,
                                             float* __restrict__ out) {
  const int f = blockIdx.x;
  const int tid = threadIdx.x;
  const int wave = tid >> 5;
  const int lane = tid & 31;
  const int n = lane & 15;
  const int h = lane >> 4;

  const float* pw = Wf + f * 32 * NE;
  v2f bf[8][2];
#pragma unroll
  for (int s = 0; s < 8; ++s)
#pragma unroll
    for (int nh = 0; nh < 2; ++nh)
      bf[s][nh] = load_bfrag(pw, 4 * s + 2 * h, n + 16 * nh);
  const float bias0 = biasF[f * 32 + n];
  const float bias1 = biasF[f * 32 + n + 16];

  const int kb = 2 * h;
  for (int t = 0; t < 8; ++t) {
    const int btile = blockIdx.y * 8 + wave + 128 * t;
    const float* prow = emb + (size_t)(btile * 16 + n) * ROW_STRIDE + f * NE;
    if (t < 7) __builtin_prefetch(prow + (size_t)128 * 16 * ROW_STRIDE, 0, 3);
    v2f a[8];
#pragma unroll
    for (int s = 0; s < 8; ++s)
      a[s] = *reinterpret_cast<const v2f*>(prow + 4 * s + kb);

    v8f c0, c1;
#pragma unroll
    for (int r = 0; r < 8; ++r) {
      c0[r] = bias0;
      c1[r] = bias1;
    }
#pragma unroll
    for (int s = 0; s < 8; ++s) {
      c0 = wmma4(a[s], bf[s][0], c0);
      c1 = wmma4(a[s], bf[s][1], c1);
    }

    // D layout: lane holds column n (+16*nh); VGPR r -> row r + 8*h
    float* po = out + (size_t)(btile * 16) * ROW_STRIDE + f * NE;
#pragma unroll
    for (int r = 0; r < 8; ++r) {
      const int row = r + 8 * h;
      po[(size_t)row * ROW_STRIDE + n] = c0[r];
      po[(size_t)row * ROW_STRIDE + n + 16] = c1[r];
    }
  }
}

// ---------------------------------------------------------------------------
extern "C" void kernel_launch(void* const* d_in, const int* in_sizes, int n_in,
                              void* d_out, int out_size, void* d_ws,
                              size_t ws_size, hipStream_t stream) {
  const float* emb = (const float*)d_in[0];
  const float* w4 = (const float*)d_in[1];
  const float* w8 = (const float*)d_in[2];
  const float* w16 = (const float*)d_in[3];
  const float* w32 = (const float*)d_in[4];
  const float* gate = (const float*)d_in[5];
  const float* noise = (const float*)d_in[6];
  float* out = (float*)d_out;

  float* ws = (float*)d_ws;
  float* sums = ws;                    // [2][4][32] = 256 floats
  float* Wf = ws + 256;                // [39][32][32] = 39936 floats
  float* biasF = ws + 256 + NF * 1024; // [39][32] = 1248 floats

  k_zero<<<1, 256, 0, stream>>>(sums);
  k_stats<<<dim3(NF, 16), 256, 0, stream>>>(emb, w4, w8, w16, w32, sums);
  k_finalize<<<40, 256, 0, stream>>>(sums, gate, noise, w4, w8, w16, w32, Wf,
                                     biasF);
  k_out<<<dim3(NF, 16), 256, 0, stream>>>(emb, Wf, biasF, out);
}